// SumAggregationLayer_v2_87574383165771
// MI455X (gfx1250) — compile-verified
//
#include <hip/hip_runtime.h>

// Segment-sum over feature axis: out[b,g] = sum_{k<16} x[b, 16g+k]
//   x: [4096, 16384] f32  ->  out: [4096, 1024] f32
// HBM-bound (~272 MiB @ 23.3 TB/s => ~12us floor). Reduction expressed as
// D = A x S via V_WMMA_F32_16X16X4_F32 (exact f32 adds), hidden under HBM time.
// Data staged memory->LDS via GLOBAL_LOAD_ASYNC_TO_LDS_B128 (ASYNCcnt) with
// per-wave double buffering; falls back to b128-load + ds_store if the async
// builtin is unavailable.

typedef __attribute__((ext_vector_type(2))) float v2f;
typedef __attribute__((ext_vector_type(8))) float v8f;
typedef __attribute__((ext_vector_type(4))) int   v4i;

#define BATCH_N      4096
#define SIZE_IN_N    16384
#define NUM_GROUPS_N 1024
#define WAVES_PER_BLOCK 8
#define STAGE_COLS   32
#define NUM_STAGES   8             // 8 * 32 = 256 cols per 16x16 output tile
#define LDS_STRIDE   36            // 32 data cols + 4 pad -> conflict-free ds_load_b64
#define BUF_FLOATS   (16 * LDS_STRIDE)

#if defined(__HIP_DEVICE_COMPILE__) && __has_builtin(__builtin_amdgcn_global_load_async_to_lds_b128)
#define HAS_ASYNC 1
#else
#define HAS_ASYNC 0
#endif

#if HAS_ASYNC
// Builtin signature (from hipcc diagnostic): param0 = v4i in addrspace(1)
// (printed "__device__"), param1 = v4i in addrspace(3), then imm offset, imm cpol.
typedef __attribute__((address_space(1))) v4i GV4;
typedef __attribute__((address_space(3))) v4i LV4;
#endif

// Stage one 16-row x 32-col f32 slab (2 KB) into a wave-private LDS buffer.
// 4 instructions/wave, each moving 32 lanes x 16 B = 512 B, fully coalesced
// (4 rows x 128 contiguous bytes per instruction).
__device__ __forceinline__ void stage_to_lds(const float* __restrict__ src,
                                             float* __restrict__ dst, int lane) {
#pragma unroll
    for (int i = 0; i < 4; ++i) {
        const int v    = i * 32 + lane;   // float4 index in 16x8 float4 slab
        const int row  = v >> 3;
        const int colv = v & 7;
        const float* gp = src + (size_t)row * SIZE_IN_N + colv * 4;
        float*       lp = dst + row * LDS_STRIDE + colv * 4;
#if HAS_ASYNC
        float* gpm = const_cast<float*>(gp);
        __builtin_amdgcn_global_load_async_to_lds_b128((GV4*)gpm, (LV4*)lp, 0, 0);
#else
        *(float4*)lp = *(const float4*)gp;
#endif
    }
}

__global__ __launch_bounds__(WAVES_PER_BLOCK * 32)
void seg_sum_wmma_kernel(const float* __restrict__ x, float* __restrict__ out) {
    __shared__ float lds_all[WAVES_PER_BLOCK * 2 * BUF_FLOATS];

    const int lane = (int)(threadIdx.x & 31);
    const int wave = (int)(threadIdx.x >> 5);
    const int wid  = (int)blockIdx.x * WAVES_PER_BLOCK + wave;

    // 256 batch tiles x 64 group tiles; one 16x16 output tile per wave.
    const int btile = wid >> 6;
    const int gtile = wid & 63;
    const size_t b0 = (size_t)btile * 16;
    const int    g0 = gtile * 16;

    const float* xt = x + b0 * SIZE_IN_N + (size_t)g0 * 16;   // 16 x 256 strip
    float* buf0 = lds_all + wave * (2 * BUF_FLOATS);          // wave-private ping
    float* buf1 = buf0 + BUF_FLOATS;                          // wave-private pong

    const int lrow  = lane & 15;   // N (group col) / A row (batch)
    const int lhalf = lane >> 4;   // K-half selector of the f32 A layout

    v8f acc = {};

    stage_to_lds(xt, buf0, lane);  // prologue: stage 0 in flight

#pragma unroll
    for (int s = 0; s < NUM_STAGES; ++s) {
        float* cur = (s & 1) ? buf1 : buf0;

        if (s < NUM_STAGES - 1) {
#if HAS_ASYNC
            if (s >= 1) {
                // Next issue overwrites the buffer whose ds_loads ran in
                // iteration s-1; async LDS writes are unordered vs DScnt ops.
                asm volatile("s_wait_dscnt 0x0" ::: "memory");
            }
#endif
            float* nxt = (s & 1) ? buf0 : buf1;
            stage_to_lds(xt + (s + 1) * STAGE_COLS, nxt, lane);
        }
#if HAS_ASYNC
        // Async loads complete in order: <=4 outstanding => stage s resident.
        if (s < NUM_STAGES - 1) asm volatile("s_wait_asynccnt 0x4" ::: "memory");
        else                    asm volatile("s_wait_asynccnt 0x0" ::: "memory");
#endif
        // 8 chunks of K=4; chunk cc feeds output column cc/4. B's four rows are
        // identical, so the result is invariant to the A-layout K permutation.
#pragma unroll
        for (int c = 0; c < 8; ++c) {
            const int   cc = s * 8 + c;
            const float bv = (lrow == (cc >> 2)) ? 1.0f : 0.0f;
            v2f bmat = {bv, bv};
            const float* ap = cur + lrow * LDS_STRIDE + c * 4 + 2 * lhalf;
            v2f amat = *(const v2f*)ap;   // ds_load_b64, all 64 banks hit once
            acc = __builtin_amdgcn_wmma_f32_16x16x4_f32(
                /*neg_a=*/false, amat, /*neg_b=*/false, bmat,
                /*c_mod=*/(short)0, acc, /*reuse_a=*/false, /*reuse_b=*/false);
        }
    }

    // C/D layout: VGPR r, lane L -> M = r + 8*(L>=16), N = L&15.
    float* obase = out + (b0 + (size_t)(8 * lhalf)) * NUM_GROUPS_N + g0 + lrow;
#pragma unroll
    for (int r = 0; r < 8; ++r) {
        obase[(size_t)r * NUM_GROUPS_N] = acc[r];
    }
}

extern "C" void kernel_launch(void* const* d_in, const int* in_sizes, int n_in,
                              void* d_out, int out_size, void* d_ws, size_t ws_size,
                              hipStream_t stream) {
    (void)in_sizes; (void)n_in; (void)out_size; (void)d_ws; (void)ws_size;
    const float* x   = (const float*)d_in[0];
    float*       out = (float*)d_out;

    const int total_waves = (BATCH_N / 16) * (NUM_GROUPS_N / 16);  // 16384
    dim3 grid(total_waves / WAVES_PER_BLOCK);                      // 2048
    dim3 block(WAVES_PER_BLOCK * 32);                              // 256
    hipLaunchKernelGGL(seg_sum_wmma_kernel, grid, block, 0, stream, x, out);
}